// LEOBlock_65309272703149
// MI455X (gfx1250) — compile-verified
//
#include <hip/hip_runtime.h>
#include <hip/hip_bf16.h>

typedef __attribute__((ext_vector_type(2))) float v2f;
typedef __attribute__((ext_vector_type(8))) float v8f;

#define S_ROWS 1000
#define FEAT   640
#define LAT    64
#define REL    128
#define NWAY   20
#define KSHOT  50

// ---------------------------------------------------------------------------
// Kernel 1: per-class mean of support rows, padded to 32 classes (zeros).
// support [1000,640] -> clsMean [32,640]
// ---------------------------------------------------------------------------
__global__ __launch_bounds__(128) void leo_class_mean(
    const float* __restrict__ support, float* __restrict__ clsMean) {
  const int c = blockIdx.x;  // 0..31
  for (int f = threadIdx.x; f < FEAT; f += blockDim.x) {
    float s = 0.0f;
    if (c < NWAY) {
      const float* p = support + (size_t)(c * KSHOT) * FEAT + f;
      #pragma unroll 5
      for (int r = 0; r < KSHOT; ++r) s += p[(size_t)r * FEAT];
      s *= (1.0f / (float)KSHOT);
    }
    clsMean[c * FEAT + f] = s;
  }
}

// ---------------------------------------------------------------------------
// Kernel 2: clsEmb[32,64] = clsMean[32,640] @ W_enc^T   (W_enc is [64,640])
// One block, 8 waves; wave w handles output tile (mt = w>>2, nt = w&3).
// WMMA f32 16x16x4, K = 640 -> 160 WMMA issues per wave.
// ---------------------------------------------------------------------------
__global__ __launch_bounds__(256) void leo_enc_gemm(
    const float* __restrict__ A,       // clsMean [32,640]
    const float* __restrict__ Wenc,    // [64,640]; B[k][n] = Wenc[n*640+k]
    float* __restrict__ C) {           // clsEmb [32,64]
  const int lane = threadIdx.x;        // 0..31
  const int wave = threadIdx.y;        // 0..7
  const int mt = wave >> 2;            // 0..1
  const int nt = wave & 3;             // 0..3
  const int m    = lane & 15;
  const int koff = (lane >> 4) << 1;   // 0 or 2
  const int arow = mt * 16 + m;        // A row for this lane
  const int ncol = nt * 16 + m;        // B column (== W_enc row)
  v8f acc = {};
  for (int k = 0; k < FEAT; k += 4) {
    v2f a, b;
    a.x = A[arow * FEAT + k + koff];
    a.y = A[arow * FEAT + k + koff + 1];
    b.x = Wenc[ncol * FEAT + k + koff];
    b.y = Wenc[ncol * FEAT + k + koff + 1];
    acc = __builtin_amdgcn_wmma_f32_16x16x4_f32(
        false, a, false, b, (short)0, acc, false, false);
  }
  const int n    = nt * 16 + (lane & 15);
  const int half = lane >> 4;
  #pragma unroll
  for (int v = 0; v < 8; ++v)
    C[(mt * 16 + half * 8 + v) * LAT + n] = acc[v];
}

// ---------------------------------------------------------------------------
// Kernel 3: C[128,128] = A[128,128] @ B[128,128] via WMMA f32 16x16x4.
// 8 blocks x 8 waves = 64 waves; one 16x16 output tile per wave, 32 WMMAs.
// ---------------------------------------------------------------------------
__global__ __launch_bounds__(256) void leo_gemm128(
    const float* __restrict__ A, const float* __restrict__ B,
    float* __restrict__ C) {
  const int lane = threadIdx.x;
  const int wg   = blockIdx.x * 8 + threadIdx.y;  // 0..63
  const int mt = wg >> 3;
  const int nt = wg & 7;
  const int m    = lane & 15;
  const int koff = (lane >> 4) << 1;
  const int arow = mt * 16 + m;
  const int bcol = nt * 16 + m;
  v8f acc = {};
  for (int k = 0; k < REL; k += 4) {
    v2f a, b;
    a.x = A[arow * REL + k + koff];
    a.y = A[arow * REL + k + koff + 1];
    b.x = B[(k + koff) * REL + bcol];
    b.y = B[(k + koff + 1) * REL + bcol];
    acc = __builtin_amdgcn_wmma_f32_16x16x4_f32(
        false, a, false, b, (short)0, acc, false, false);
  }
  const int n    = nt * 16 + (lane & 15);
  const int half = lane >> 4;
  #pragma unroll
  for (int v = 0; v < 8; ++v)
    C[(mt * 16 + half * 8 + v) * REL + n] = acc[v];
}

// ---------------------------------------------------------------------------
// Kernel 4: codes_core[c][n] = clsEmb[c]·M[n][0:64] + meanEmb·M[n][64:128],
// then broadcast each value across the 50 shots of its class.
// out is [20,50,128].
// ---------------------------------------------------------------------------
__global__ __launch_bounds__(256) void leo_finalize(
    const float* __restrict__ clsEmb,  // [32,64], rows 0..19 valid
    const float* __restrict__ Mmat,    // [128,128] = W3@W2@W1
    float* __restrict__ out) {         // [20,50,128]
  __shared__ float meanEmb[LAT];
  __shared__ float bias[REL];
  const int tid = threadIdx.x;
  if (tid < LAT) {
    float s = 0.0f;
    #pragma unroll
    for (int c = 0; c < NWAY; ++c) s += clsEmb[c * LAT + tid];
    meanEmb[tid] = s * (1.0f / (float)NWAY);
  }
  __syncthreads();
  if (tid < REL) {
    float s = 0.0f;
    #pragma unroll 8
    for (int k = 0; k < LAT; ++k) s += meanEmb[k] * Mmat[tid * REL + LAT + k];
    bias[tid] = s;
  }
  __syncthreads();
  for (int idx = tid; idx < NWAY * REL; idx += 256) {
    const int c = idx >> 7;
    const int n = idx & (REL - 1);
    float s = bias[n];
    #pragma unroll 8
    for (int k = 0; k < LAT; ++k) s += clsEmb[c * LAT + k] * Mmat[n * REL + k];
    float* o = out + (size_t)(c * KSHOT) * REL + n;
    #pragma unroll 5
    for (int sh = 0; sh < KSHOT; ++sh) o[(size_t)sh * REL] = s;
  }
}

// ---------------------------------------------------------------------------
// Launch: classMean -> encGEMM -> (W2@W1) -> W3@(W2@W1) -> finalize+broadcast
// ---------------------------------------------------------------------------
extern "C" void kernel_launch(void* const* d_in, const int* in_sizes, int n_in,
                              void* d_out, int out_size, void* d_ws, size_t ws_size,
                              hipStream_t stream) {
  const float* support = (const float*)d_in[0];  // [1000,640]
  const float* W_enc   = (const float*)d_in[1];  // [64,640]
  const float* W1      = (const float*)d_in[2];  // [128,128]
  const float* W2      = (const float*)d_in[3];  // [128,128]
  const float* W3      = (const float*)d_in[4];  // [128,128]
  float* out = (float*)d_out;                    // [20,50,128]

  float* ws = (float*)d_ws;
  float* clsMean = ws;                 // 32*640  = 20480 floats
  float* clsEmb  = clsMean + 32 * FEAT;// 32*64   = 2048
  float* T1      = clsEmb + 32 * LAT;  // 128*128 = 16384
  float* Mmat    = T1 + REL * REL;     // 128*128 = 16384

  leo_class_mean<<<32, 128, 0, stream>>>(support, clsMean);
  leo_enc_gemm<<<1, dim3(32, 8), 0, stream>>>(clsMean, W_enc, clsEmb);
  leo_gemm128<<<8, dim3(32, 8), 0, stream>>>(W2, W1, T1);      // T1 = W2@W1
  leo_gemm128<<<8, dim3(32, 8), 0, stream>>>(W3, T1, Mmat);    // M  = W3@T1
  leo_finalize<<<1, 256, 0, stream>>>(clsEmb, Mmat, out);
}